// GIN_89945205112918
// MI455X (gfx1250) — compile-verified
//
#include <hip/hip_runtime.h>

typedef float v2f __attribute__((ext_vector_type(2)));
typedef float v8f __attribute__((ext_vector_type(8)));

#define NNODES 50000
#define NEDGES 800000
#define HDIM   64
#define NB     50
#define NL     7

// LDS weight tile: stored transposed, sWt[n][k], padded k-stride of 68 so the
// per-lane ds_load_b64 of (k, k+1) is bank-conflict-free across lanes.
#define WSTRIDE 68

// ---------------------------------------------------------------------------
// Build h0 = [x | pos | zeros] as an N x 64 row-major matrix (IN=32 padded).
// ---------------------------------------------------------------------------
__global__ void gin_build_h0(const float* __restrict__ x,
                             const float* __restrict__ pos,
                             float* __restrict__ h) {
  int t = blockIdx.x * blockDim.x + threadIdx.x;
  int node = t >> 6;
  int f = t & 63;
  if (node >= NNODES) return;
  float v = 0.0f;
  if (f < 29)      v = x[node * 29 + f];
  else if (f < 32) v = pos[node * 3 + (f - 29)];
  h[node * HDIM + f] = v;
}

// ---------------------------------------------------------------------------
// Zero-pad W1_first (32x64) to a 64x64 matrix (rows 32..63 = 0).
// ---------------------------------------------------------------------------
__global__ void gin_pad_w1(const float* __restrict__ w1, float* __restrict__ wp) {
  int t = blockIdx.x * blockDim.x + threadIdx.x;
  if (t >= 64 * 64) return;
  int k = t >> 6;
  wp[t] = (k < 32) ? w1[t] : 0.0f;
}

// ---------------------------------------------------------------------------
// Edge scatter-add: agg[dst[e]][:] += h[src[e]][:].
// One wave32 per edge: e is wave-uniform -> readfirstlane forces the index
// fetches into scalar loads (s_load) once per wave. Each lane handles a
// float2 -> 64 hardware f32 atomics per edge (the semantic minimum).
// ---------------------------------------------------------------------------
__global__ void gin_scatter_add(const int* __restrict__ src,
                                const int* __restrict__ dst,
                                const float* __restrict__ h,
                                float* __restrict__ agg) {
  int t = blockIdx.x * blockDim.x + threadIdx.x;
  int e = t >> 5;                 // wave-uniform (32 consecutive tids per wave)
  if (e >= NEDGES) return;
  e = __builtin_amdgcn_readfirstlane(e);
  int lane = threadIdx.x & 31;
  int s = src[e];                 // scalar load
  int d = dst[e];                 // scalar load
  v2f v = *(const v2f*)(h + s * HDIM + lane * 2);
  float* p = agg + d * HDIM + lane * 2;
  unsafeAtomicAdd(p,     v.x);
  unsafeAtomicAdd(p + 1, v.y);
}

// ---------------------------------------------------------------------------
// Fused GIN MLP half-layer: out = act((A [+ A2]) @ W + bias)
// A: N x 64 row-major, W: 64 x 64 row-major (K x N), out: N x 64.
//
// One wave32 per 16x16 output tile; K=64 via 16 chained V_WMMA_F32_16X16X4_F32.
// The 8 waves of a block cover 2 row-tiles x all 4 col-tiles, so the whole W
// is staged once per block in LDS (transposed, padded) and B fragments come
// from conflict-free ds_load_b64. A and B fragments are preloaded into
// register arrays so the WMMA chain issues back-to-back.
//
// ISA operand layouts (CDNA5 ISA):
//   A 16x4 f32 : lanes 0-15 -> M=lane, K={4k,4k+1}; lanes 16-31 -> K={4k+2,4k+3}
//   B 4x16 f32 : lanes 0-15 -> N=lane, K={4k,4k+1}; lanes 16-31 -> K={4k+2,4k+3}
//   C/D        : VGPR i -> M = i + 8*(lane>>4), N = lane&15
// ---------------------------------------------------------------------------
template <bool FUSE_ADD>
__global__ void gin_mlp_gemm(const float* __restrict__ A,
                             const float* __restrict__ A2,
                             const float* __restrict__ W,
                             const float* __restrict__ bias,
                             float* __restrict__ out,
                             int doRelu) {
  __shared__ float sWt[HDIM * WSTRIDE];  // transposed: sWt[n*WSTRIDE + k]

  // Cooperative staging of W (64x64) into LDS, transposed.
  for (int i = threadIdx.x; i < HDIM * HDIM; i += blockDim.x) {
    int k = i >> 6;
    int n = i & 63;
    sWt[n * WSTRIDE + k] = W[i];
  }
  __syncthreads();

  int wave = threadIdx.x >> 5;
  int lane = threadIdx.x & 31;
  int tile = blockIdx.x * (blockDim.x >> 5) + wave;
  const int nTiles = (NNODES / 16) * (HDIM / 16);  // 3125 * 4 = 12500
  if (tile >= nTiles) return;                      // wave-uniform exit

  int rowTile = tile >> 2;
  int colTile = tile & 3;
  int half = lane >> 4;
  int l15  = lane & 15;

  const float* aRow = A + (size_t)(rowTile * 16 + l15) * HDIM;
  int bCol = colTile * 16 + l15;

  // Preload A fragments (one load clause, one wait).
  v2f aF[16];
#pragma unroll
  for (int k = 0; k < 16; ++k) {
    aF[k] = *(const v2f*)(aRow + 4 * k + 2 * half);
  }
  if (FUSE_ADD) {
    const float* a2Row = A2 + (size_t)(rowTile * 16 + l15) * HDIM;
#pragma unroll
    for (int k = 0; k < 16; ++k) {
      v2f t = *(const v2f*)(a2Row + 4 * k + 2 * half);
      aF[k].x += t.x;
      aF[k].y += t.y;
    }
  }

  // Preload B fragments from LDS: (k, k+1) pair is one b64 load.
  v2f bF[16];
#pragma unroll
  for (int k = 0; k < 16; ++k) {
    bF[k] = *(const v2f*)(&sWt[bCol * WSTRIDE + 4 * k + 2 * half]);
  }

  // Back-to-back WMMA accumulation chain (D->C accumulation, no A/B hazard).
  v8f acc = {};
#pragma unroll
  for (int k = 0; k < 16; ++k) {
    acc = __builtin_amdgcn_wmma_f32_16x16x4_f32(
        /*neg_a=*/false, aF[k], /*neg_b=*/false, bF[k],
        /*c_mod=*/(short)0, acc, /*reuse_a=*/false, /*reuse_b=*/false);
  }

  float bv = bias[bCol];
#pragma unroll
  for (int i = 0; i < 8; ++i) {
    int r = rowTile * 16 + i + 8 * half;
    float v = acc[i] + bv;
    if (doRelu) v = fmaxf(v, 0.0f);
    out[(size_t)r * HDIM + bCol] = v;
  }
}

// ---------------------------------------------------------------------------
// Global mean pool accumulation: one wave32 per node; batch[n] via scalar load.
// ---------------------------------------------------------------------------
__global__ void gin_pool_accum(const float* __restrict__ h,
                               const int* __restrict__ batch,
                               float* __restrict__ sums,
                               float* __restrict__ counts) {
  int t = blockIdx.x * blockDim.x + threadIdx.x;
  int n = t >> 5;                 // wave-uniform
  if (n >= NNODES) return;
  n = __builtin_amdgcn_readfirstlane(n);
  int lane = threadIdx.x & 31;
  int b = batch[n];               // scalar load
  v2f v = *(const v2f*)(h + (size_t)n * HDIM + lane * 2);
  float* p = sums + b * HDIM + lane * 2;
  unsafeAtomicAdd(p,     v.x);
  unsafeAtomicAdd(p + 1, v.y);
  if (lane == 0) unsafeAtomicAdd(&counts[b], 1.0f);
}

// ---------------------------------------------------------------------------
// out[b] = (sums[b]/max(counts[b],1)) . lin_w + lin_b     (B=50, tiny)
// ---------------------------------------------------------------------------
__global__ void gin_final_linear(const float* __restrict__ sums,
                                 const float* __restrict__ counts,
                                 const float* __restrict__ lin_w,
                                 const float* __restrict__ lin_b,
                                 float* __restrict__ out) {
  int b = blockIdx.x * blockDim.x + threadIdx.x;
  if (b >= NB) return;
  float inv = 1.0f / fmaxf(counts[b], 1.0f);
  float acc = 0.0f;
#pragma unroll
  for (int f = 0; f < HDIM; ++f) acc += sums[b * HDIM + f] * inv * lin_w[f];
  out[b] = acc + lin_b[0];
}

// ---------------------------------------------------------------------------
extern "C" void kernel_launch(void* const* d_in, const int* in_sizes, int n_in,
                              void* d_out, int out_size, void* d_ws, size_t ws_size,
                              hipStream_t stream) {
  const float* x     = (const float*)d_in[0];
  const float* pos   = (const float*)d_in[1];
  const int*   ei    = (const int*)d_in[2];   // (2, E): row 0 = src, row 1 = dst
  const int*   batch = (const int*)d_in[3];
  const float* W1f   = (const float*)d_in[4]; // (32, 64)
  const float* W1r   = (const float*)d_in[5]; // (L-1, 64, 64)
  const float* b1    = (const float*)d_in[6]; // (L, 64)
  const float* W2    = (const float*)d_in[7]; // (L, 64, 64)
  const float* b2    = (const float*)d_in[8]; // (L, 64)
  const float* lw    = (const float*)d_in[9]; // (64, 1)
  const float* lb    = (const float*)d_in[10];
  float* out = (float*)d_out;

  // Workspace layout (bytes); whole working set is L2-resident on MI455X.
  const size_t NH = (size_t)NNODES * HDIM * sizeof(float);  // 12.8 MB
  char* ws = (char*)d_ws;
  float* h    = (float*)(ws);              // current node features  [N,64]
  float* agg  = (float*)(ws + NH);         // neighborhood sums      [N,64]
  float* ybuf = (float*)(ws + 2 * NH);     // MLP hidden             [N,64]
  float* Wp   = (float*)(ws + 3 * NH);                 // padded W1  [64,64]
  float* sums = (float*)(ws + 3 * NH + 64 * 64 * 4);   // pool sums  [B,64]
  float* cnts = sums + NB * HDIM;                      // pool counts [B]

  const int* src = ei;
  const int* dst = ei + NEDGES;

  const int gemmBlocks = ((NNODES / 16) * (HDIM / 16) + 7) / 8;  // 8 waves/block

  gin_build_h0<<<(NNODES * HDIM + 255) / 256, 256, 0, stream>>>(x, pos, h);
  gin_pad_w1<<<(64 * 64 + 255) / 256, 256, 0, stream>>>(W1f, Wp);

  for (int l = 0; l < NL; ++l) {
    hipMemsetAsync(agg, 0, NH, stream);
    gin_scatter_add<<<(NEDGES * 32) / 256, 256, 0, stream>>>(src, dst, h, agg);

    const float* Wa = (l == 0) ? Wp : (W1r + (size_t)(l - 1) * HDIM * HDIM);
    // y = relu((h + agg) @ Wa + b1[l])
    gin_mlp_gemm<true><<<gemmBlocks, 256, 0, stream>>>(h, agg, Wa, b1 + l * HDIM,
                                                       ybuf, 1);
    // h = act(y @ W2[l] + b2[l]); ReLU after every layer except the last
    gin_mlp_gemm<false><<<gemmBlocks, 256, 0, stream>>>(ybuf, nullptr,
                                                        W2 + (size_t)l * HDIM * HDIM,
                                                        b2 + l * HDIM, h,
                                                        (l < NL - 1) ? 1 : 0);
  }

  hipMemsetAsync(sums, 0, (NB * HDIM + NB) * sizeof(float), stream);
  gin_pool_accum<<<(NNODES * 32 + 255) / 256, 256, 0, stream>>>(h, batch, sums, cnts);
  gin_final_linear<<<1, 64, 0, stream>>>(sums, cnts, lw, lb, out);
}